// SetAbstraction_75024488726596
// MI455X (gfx1250) — compile-verified
//
#include <hip/hip_runtime.h>
#include <math.h>

#define NPTS   16384
#define MCTR   4096
#define KNB    32
#define CINF   35          // 3 + CIN
#define CPAD   36          // padded K for GEMM1
#define CMID   32
#define COUT   64
#define NROW   (MCTR*KNB)  // 131072 rows through both GEMMs
#define BN_EPS 1e-5f

typedef float v2f __attribute__((ext_vector_type(2)));
typedef float v8f __attribute__((ext_vector_type(8)));

static __device__ __forceinline__ v8f v8zero() {
    v8f z;
#pragma unroll
    for (int i = 0; i < 8; ++i) z[i] = 0.0f;
    return z;
}

// ---------------------------------------------------------------------------
// 0) prep: pad W1 (35x32) -> W1p (36x32), zero last row
// ---------------------------------------------------------------------------
__global__ __launch_bounds__(256) void prep_kernel(const float* __restrict__ W1,
                                                   float* __restrict__ W1p) {
    int i = blockIdx.x * 256 + threadIdx.x;
    if (i < CPAD * CMID) {
        int r = i >> 5;
        W1p[i] = (r < CINF) ? W1[i] : 0.0f;  // row-major, identical layout for r<35
    }
}

// ---------------------------------------------------------------------------
// 1) Furthest point sampling: single 1024-thread block, points in registers
// ---------------------------------------------------------------------------
__global__ __launch_bounds__(1024) void fps_kernel(const float* __restrict__ p,
                                                   int* __restrict__ fidx) {
    const int t = threadIdx.x;  // 0..1023, each owns 16 points (id = j*1024+t)
    float px[16], py[16], pz[16], dd[16];
#pragma unroll
    for (int j = 0; j < 16; ++j) {
        int id = j * 1024 + t;
        px[j] = p[id * 3 + 0];
        py[j] = p[id * 3 + 1];
        pz[j] = p[id * 3 + 2];
        dd[j] = 1e10f;
    }
    __shared__ float rv[32];
    __shared__ int   ri[32];
    __shared__ int   sfar;

    int far = 0;
    for (int m = 0; m < MCTR; ++m) {
        if (t == 0) fidx[m] = far;
        float fx = p[far * 3 + 0];
        float fy = p[far * 3 + 1];
        float fz = p[far * 3 + 2];
        float bv = -1.0f;
        int   bi = 0x7fffffff;
#pragma unroll
        for (int j = 0; j < 16; ++j) {
            float dx = px[j] - fx, dy = py[j] - fy, dz = pz[j] - fz;
            float d  = dx * dx + dy * dy + dz * dz;
            float nd = fminf(dd[j], d);
            dd[j] = nd;
            int id = j * 1024 + t;
            if (nd > bv || (nd == bv && id < bi)) { bv = nd; bi = id; }
        }
        // wave32 argmax (first-max tie-break = lowest index)
#pragma unroll
        for (int off = 16; off; off >>= 1) {
            float ov = __shfl_down(bv, off, 32);
            int   oi = __shfl_down(bi, off, 32);
            if (ov > bv || (ov == bv && oi < bi)) { bv = ov; bi = oi; }
        }
        int w = t >> 5;
        if ((t & 31) == 0) { rv[w] = bv; ri[w] = bi; }
        __syncthreads();
        if (w == 0) {
            bv = rv[t & 31];
            bi = ri[t & 31];
#pragma unroll
            for (int off = 16; off; off >>= 1) {
                float ov = __shfl_down(bv, off, 32);
                int   oi = __shfl_down(bi, off, 32);
                if (ov > bv || (ov == bv && oi < bi)) { bv = ov; bi = oi; }
            }
            if (t == 0) sfar = bi;
        }
        __syncthreads();
        far = sfar;
    }
}

// ---------------------------------------------------------------------------
// 2) Ball query + gather: one wave per center; first K hits in ascending index
//    order; writes new_p into d_out[0..3M) and padded feature rows [MCTR*KNB,36]
// ---------------------------------------------------------------------------
__global__ __launch_bounds__(32) void ballquery_gather_kernel(
    const float* __restrict__ p, const float* __restrict__ x,
    const int* __restrict__ fidx, float* __restrict__ feat,
    float* __restrict__ newp_out) {
    const int c    = blockIdx.x;
    const int lane = threadIdx.x;
    const int ci   = fidx[c];
    const float cx = p[ci * 3 + 0];
    const float cy = p[ci * 3 + 1];
    const float cz = p[ci * 3 + 2];
    if (lane < 3) newp_out[c * 3 + lane] = p[ci * 3 + lane];

    __shared__ int sidx[KNB];
    const float r2 = 0.1f * 0.1f;
    int cnt = 0;
    for (int base = 0; base < NPTS && cnt < KNB; base += 32) {
        int j = base + lane;
        float dx = p[j * 3 + 0] - cx;
        float dy = p[j * 3 + 1] - cy;
        float dz = p[j * 3 + 2] - cz;
        float d2 = dx * dx + dy * dy + dz * dz;
        int hit = (d2 <= r2) ? 1 : 0;
        unsigned mask = (unsigned)__ballot(hit);
        int prefix = __popc(mask & ((1u << lane) - 1u));
        if (hit && (cnt + prefix) < KNB) sidx[cnt + prefix] = j;
        cnt += __popc(mask);
        if (cnt > KNB) cnt = KNB;
    }
    __syncthreads();
    int nb = ci;
    if (cnt > 0) nb = (lane < cnt) ? sidx[lane] : sidx[0];

    float* f = feat + (size_t)(c * KNB + lane) * CPAD;
    f[0] = p[nb * 3 + 0] - cx;
    f[1] = p[nb * 3 + 1] - cy;
    f[2] = p[nb * 3 + 2] - cz;
#pragma unroll
    for (int q = 0; q < 32; ++q) f[3 + q] = x[nb * 32 + q];
    f[35] = 0.0f;
}

// ---------------------------------------------------------------------------
// 3) GEMM1: [NROW,36] @ W1p[36,32] + b1 -> h1 (raw). One wave per 16-row strip,
//    exact-f32 WMMA 16x16x4, 9 K-steps, 2 column tiles.
// ---------------------------------------------------------------------------
__global__ __launch_bounds__(32) void gemm1_kernel(const float* __restrict__ feat,
                                                   const float* __restrict__ W1p,
                                                   const float* __restrict__ b1,
                                                   float* __restrict__ h1) {
    const int strip = blockIdx.x;        // 0..NROW/16-1
    const int lane  = threadIdx.x;
    const int col   = lane & 15;
    const int khalf = (lane >> 4) * 2;   // 0 or 2 per ISA A/B layout
    const int arow  = strip * 16 + col;  // A row held by this lane
    const float* a_ptr = feat + (size_t)arow * CPAD;

    v8f acc0 = v8zero();
    v8f acc1 = v8zero();
#pragma unroll
    for (int k0 = 0; k0 < CPAD; k0 += 4) {
        v2f a;
        a.x = a_ptr[k0 + khalf];
        a.y = a_ptr[k0 + khalf + 1];
        v2f bA, bB;
        bA.x = W1p[(k0 + khalf) * CMID + col];
        bA.y = W1p[(k0 + khalf + 1) * CMID + col];
        bB.x = W1p[(k0 + khalf) * CMID + col + 16];
        bB.y = W1p[(k0 + khalf + 1) * CMID + col + 16];
        acc0 = __builtin_amdgcn_wmma_f32_16x16x4_f32(false, a, false, bA, (short)0, acc0, false, false);
        acc1 = __builtin_amdgcn_wmma_f32_16x16x4_f32(false, a, false, bB, (short)0, acc1, false, false);
    }
    const int rbase = strip * 16 + ((lane >> 4) ? 8 : 0);
    const float biasA = b1[col];
    const float biasB = b1[col + 16];
#pragma unroll
    for (int i = 0; i < 8; ++i) {
        h1[(size_t)(rbase + i) * CMID + col]      = acc0[i] + biasA;
        h1[(size_t)(rbase + i) * CMID + col + 16] = acc1[i] + biasB;
    }
}

// ---------------------------------------------------------------------------
// 4) BN1 stats: deterministic fixed-order reduction, one block per channel
// ---------------------------------------------------------------------------
__global__ __launch_bounds__(256) void bnstats1_kernel(const float* __restrict__ h1,
                                                       float* __restrict__ sum1,
                                                       float* __restrict__ ss1) {
    const int ch = blockIdx.x;   // 0..31
    const int t  = threadIdx.x;
    float s = 0.0f, q = 0.0f;
    for (int i = t; i < NROW; i += 256) {
        float v = h1[(size_t)i * CMID + ch];
        s += v;
        q += v * v;
    }
    __shared__ float sh_s[256], sh_q[256];
    sh_s[t] = s; sh_q[t] = q;
    __syncthreads();
    for (int off = 128; off; off >>= 1) {
        if (t < off) { sh_s[t] += sh_s[t + off]; sh_q[t] += sh_q[t + off]; }
        __syncthreads();
    }
    if (t == 0) { sum1[ch] = sh_s[0]; ss1[ch] = sh_q[0]; }
}

// ---------------------------------------------------------------------------
// 5) BN1 + ReLU in place
// ---------------------------------------------------------------------------
__global__ __launch_bounds__(256) void bnrelu1_kernel(float* __restrict__ h1,
                                                      const float* __restrict__ sum1,
                                                      const float* __restrict__ ss1,
                                                      const float* __restrict__ g1,
                                                      const float* __restrict__ be1) {
    const int i  = blockIdx.x * 256 + threadIdx.x;  // NROW*CMID threads exactly
    const int ch = i & 31;
    const float inv = 1.0f / (float)NROW;
    float mu  = sum1[ch] * inv;
    float var = ss1[ch] * inv - mu * mu;
    float v = (h1[i] - mu) * rsqrtf(var + BN_EPS) * g1[ch] + be1[ch];
    h1[i] = fmaxf(v, 0.0f);
}

// ---------------------------------------------------------------------------
// GEMM2 core: one wave per center computes all 32x64 raw h2 values into
// acc[strip][coltile]; exact-f32 WMMA, K=32 -> 8 steps, 64 WMMAs per wave.
// ---------------------------------------------------------------------------
static __device__ __forceinline__ void gemm2_accum(const float* __restrict__ h1,
                                                   const float* __restrict__ W2,
                                                   int c, int lane, v8f acc[2][4]) {
    const int col   = lane & 15;
    const int khalf = (lane >> 4) * 2;
#pragma unroll
    for (int s = 0; s < 2; ++s)
#pragma unroll
        for (int t = 0; t < 4; ++t) acc[s][t] = v8zero();

#pragma unroll
    for (int k0 = 0; k0 < CMID; k0 += 4) {
        v2f b[4];
#pragma unroll
        for (int t = 0; t < 4; ++t) {
            b[t].x = W2[(k0 + khalf) * COUT + t * 16 + col];
            b[t].y = W2[(k0 + khalf + 1) * COUT + t * 16 + col];
        }
#pragma unroll
        for (int s = 0; s < 2; ++s) {
            const int row = c * KNB + s * 16 + col;
            v2f a;
            a.x = h1[(size_t)row * CMID + k0 + khalf];
            a.y = h1[(size_t)row * CMID + k0 + khalf + 1];
#pragma unroll
            for (int t = 0; t < 4; ++t)
                acc[s][t] = __builtin_amdgcn_wmma_f32_16x16x4_f32(false, a, false, b[t], (short)0, acc[s][t], false, false);
        }
    }
}

// 6) GEMM2 stats pass: per-center partial sums (deterministic order)
__global__ __launch_bounds__(32) void gemm2_stats_kernel(const float* __restrict__ h1,
                                                         const float* __restrict__ W2,
                                                         const float* __restrict__ b2,
                                                         float* __restrict__ part2) {
    const int c    = blockIdx.x;
    const int lane = threadIdx.x;
    const int col  = lane & 15;
    v8f acc[2][4];
    gemm2_accum(h1, W2, c, lane, acc);
#pragma unroll
    for (int t = 0; t < 4; ++t) {
        const int ct = t * 16 + col;
        const float bias = b2[ct];
        float s = 0.0f, q = 0.0f;
#pragma unroll
        for (int st = 0; st < 2; ++st)
#pragma unroll
            for (int i = 0; i < 8; ++i) {
                float v = acc[st][t][i] + bias;
                s += v;
                q += v * v;
            }
        s += __shfl_xor(s, 16, 32);
        q += __shfl_xor(q, 16, 32);
        if (lane < 16) {
            part2[(size_t)c * 128 + ct]      = s;
            part2[(size_t)c * 128 + 64 + ct] = q;
        }
    }
}

// 7) reduce partials -> sum2/ss2, one block per channel (fixed order)
__global__ __launch_bounds__(256) void reduce2_kernel(const float* __restrict__ part2,
                                                      float* __restrict__ sum2,
                                                      float* __restrict__ ss2) {
    const int ch = blockIdx.x;   // 0..63
    const int t  = threadIdx.x;
    float s = 0.0f, q = 0.0f;
    for (int cidx = t; cidx < MCTR; cidx += 256) {
        s += part2[(size_t)cidx * 128 + ch];
        q += part2[(size_t)cidx * 128 + 64 + ch];
    }
    __shared__ float sh_s[256], sh_q[256];
    sh_s[t] = s; sh_q[t] = q;
    __syncthreads();
    for (int off = 128; off; off >>= 1) {
        if (t < off) { sh_s[t] += sh_s[t + off]; sh_q[t] += sh_q[t + off]; }
        __syncthreads();
    }
    if (t == 0) { sum2[ch] = sh_s[0]; ss2[ch] = sh_q[0]; }
}

// 8) GEMM2 fused: recompute + BN2 + ReLU + max over K neighbors -> d_out
__global__ __launch_bounds__(32) void gemm2_fused_kernel(const float* __restrict__ h1,
                                                         const float* __restrict__ W2,
                                                         const float* __restrict__ b2,
                                                         const float* __restrict__ sum2,
                                                         const float* __restrict__ ss2,
                                                         const float* __restrict__ g2,
                                                         const float* __restrict__ be2,
                                                         float* __restrict__ out) {
    const int c    = blockIdx.x;
    const int lane = threadIdx.x;
    const int col  = lane & 15;
    v8f acc[2][4];
    gemm2_accum(h1, W2, c, lane, acc);
    const float inv = 1.0f / (float)NROW;
#pragma unroll
    for (int t = 0; t < 4; ++t) {
        const int ct = t * 16 + col;
        const float bias = b2[ct];
        float mu  = sum2[ct] * inv;
        float var = ss2[ct] * inv - mu * mu;
        float sc  = rsqrtf(var + BN_EPS) * g2[ct];
        float sh  = be2[ct];
        float m = 0.0f;  // ReLU outputs are >= 0, so max >= 0
#pragma unroll
        for (int st = 0; st < 2; ++st)
#pragma unroll
            for (int i = 0; i < 8; ++i) {
                float v = fmaxf((acc[st][t][i] + bias - mu) * sc + sh, 0.0f);
                m = fmaxf(m, v);
            }
        m = fmaxf(m, __shfl_xor(m, 16, 32));
        if (lane < 16) out[(size_t)c * COUT + ct] = m;
    }
}

// ---------------------------------------------------------------------------
extern "C" void kernel_launch(void* const* d_in, const int* in_sizes, int n_in,
                              void* d_out, int out_size, void* d_ws, size_t ws_size,
                              hipStream_t stream) {
    const float* p   = (const float*)d_in[0];
    const float* x   = (const float*)d_in[1];
    const float* W1  = (const float*)d_in[2];
    const float* b1  = (const float*)d_in[3];
    const float* g1  = (const float*)d_in[4];
    const float* be1 = (const float*)d_in[5];
    const float* W2  = (const float*)d_in[6];
    const float* b2  = (const float*)d_in[7];
    const float* g2  = (const float*)d_in[8];
    const float* be2 = (const float*)d_in[9];
    float* out = (float*)d_out;   // [M*3 new_p | M*64 pooled]

    // workspace layout (4-byte units)
    float* ws    = (float*)d_ws;
    float* W1p   = ws;                               // 1152
    float* sum1  = W1p + CPAD * CMID;                // 32
    float* ss1   = sum1 + CMID;                      // 32
    float* sum2  = ss1 + CMID;                       // 64
    float* ss2   = sum2 + COUT;                      // 64
    int*   fidx  = (int*)(ss2 + COUT);               // 4096
    float* part2 = (float*)(fidx + MCTR);            // 4096*128
    float* feat  = part2 + (size_t)MCTR * 128;       // NROW*36
    float* h1    = feat + (size_t)NROW * CPAD;       // NROW*32

    prep_kernel<<<(CPAD * CMID + 255) / 256, 256, 0, stream>>>(W1, W1p);
    fps_kernel<<<1, 1024, 0, stream>>>(p, fidx);
    ballquery_gather_kernel<<<MCTR, 32, 0, stream>>>(p, x, fidx, feat, out);
    gemm1_kernel<<<NROW / 16, 32, 0, stream>>>(feat, W1p, b1, h1);
    bnstats1_kernel<<<CMID, 256, 0, stream>>>(h1, sum1, ss1);
    bnrelu1_kernel<<<(NROW * CMID) / 256, 256, 0, stream>>>(h1, sum1, ss1, g1, be1);
    gemm2_stats_kernel<<<MCTR, 32, 0, stream>>>(h1, W2, b2, part2);
    reduce2_kernel<<<COUT, 256, 0, stream>>>(part2, sum2, ss2);
    gemm2_fused_kernel<<<MCTR, 32, 0, stream>>>(h1, W2, b2, sum2, ss2, g2, be2,
                                                out + (size_t)MCTR * 3);
}